// ChamferDistanceLoss_11063835755244
// MI455X (gfx1250) — compile-verified
//
#include <hip/hip_runtime.h>

typedef float v2f __attribute__((ext_vector_type(2)));
typedef float v8f __attribute__((ext_vector_type(8)));

#define BATCH 4
#define NPTS 4096
#define TILE 16
#define WAVES_PER_BLOCK 4
#define THREADS (WAVES_PER_BLOCK * 32)

// One wave owns a 16-row tile of P and scans all 4096 columns of Q (staged in
// LDS), producing per-row minima of the squared-distance matrix.
//
// Trick: V_WMMA_F32_16X16X4_F32 has K=4 but the dot product needs only K=3.
// We load A = [-2px, -2py, -2pz, 1.0] and B = [qx, qy, qz, |q|^2], so one
// WMMA with C=0 (inline constant) yields  -2*p.q + |q|^2  per element.
// The |p|^2 term is column-invariant, so it is added once AFTER the min scan.
__global__ __launch_bounds__(THREADS)
void chamfer_rowmin_wmma(const float* __restrict__ P,   // [B, N, 3] rows
                         const float* __restrict__ Q,   // [B, M, 3] cols
                         float* __restrict__ rowmin)    // [B, N]
{
    __shared__ float sQ[NPTS * 3];                      // 48 KB of the WGP's 320 KB LDS
    const int b   = blockIdx.y;
    const int tid = threadIdx.x;

    // Stage the full column point set for this batch into LDS once.
    const float* Qb = Q + (size_t)b * NPTS * 3;
    for (int i = tid; i < NPTS * 3; i += THREADS) sQ[i] = Qb[i];
    __syncthreads();

    const int wave = tid >> 5;
    const int lane = tid & 31;
    const int lo   = lane & 15;   // N (and A-row M) position within half-wave
    const int hi   = lane >> 4;   // selects K pair {0,1} vs {2,3}; C rows +8
    const int n0   = (blockIdx.x * WAVES_PER_BLOCK + wave) * TILE;

    const float* Pb = P + (size_t)b * NPTS * 3;

    // A-matrix 16x4: lane l holds M = l&15, VGPR v holds K = v + 2*(l>>4).
    // K=0..2: -2 * coords, K=3: 1.0 (applies B's |q|^2 row).
    const int ra = n0 + lo;
    const float px = Pb[ra * 3 + 0], py = Pb[ra * 3 + 1], pz = Pb[ra * 3 + 2];
    v2f a;
    a.x = hi ? (-2.0f * pz) : (-2.0f * px);
    a.y = hi ? 1.0f         : (-2.0f * py);

    // |p|^2 for the 8 C-rows this lane accumulates (M = j + 8*hi); uniform
    // across the 16 lanes of each half-wave, applied after the reduction.
    float aa8[8];
    #pragma unroll
    for (int j = 0; j < 8; ++j) {
        const int r = n0 + j + 8 * hi;
        const float x = Pb[r * 3 + 0], y = Pb[r * 3 + 1], z = Pb[r * 3 + 2];
        aa8[j] = x * x + y * y + z * z;
    }

    float rmin[8];
    #pragma unroll
    for (int j = 0; j < 8; ++j) rmin[j] = 3.402823466e+38f;

    #pragma unroll 4
    for (int m0 = 0; m0 < NPTS; m0 += TILE) {
        const int c  = m0 + lo;
        const float qx = sQ[c * 3 + 0], qy = sQ[c * 3 + 1], qz = sQ[c * 3 + 2];
        const float bb = qx * qx + qy * qy + qz * qz;

        // B-matrix 4x16: lane l holds N = l&15, VGPR v holds K = v + 2*(l>>4).
        // Row K=3 carries |q|^2.
        v2f bq;
        bq.x = hi ? qz : qx;
        bq.y = hi ? bb : qy;

        const v8f czero = {};
        // e = -2*p.q + |q|^2  (16x16 tile), C is inline 0
        v8f e = __builtin_amdgcn_wmma_f32_16x16x4_f32(
            /*neg_a=*/false, a, /*neg_b=*/false, bq,
            /*c_mod=*/(short)0, czero, /*reuse_a=*/false, /*reuse_b=*/false);

        #pragma unroll
        for (int j = 0; j < 8; ++j) rmin[j] = fminf(rmin[j], e[j]);
    }

    // Min-reduce over the 16 lanes that hold N=0..15 (xor of bits 0..3 never
    // crosses the half-wave boundary, so both halves reduce independently).
    #pragma unroll
    for (int j = 0; j < 8; ++j) {
        float v = rmin[j];
        v = fminf(v, __shfl_xor(v, 1, 32));
        v = fminf(v, __shfl_xor(v, 2, 32));
        v = fminf(v, __shfl_xor(v, 4, 32));
        v = fminf(v, __shfl_xor(v, 8, 32));
        rmin[j] = v;
    }
    if (lo == 0) {
        #pragma unroll
        for (int j = 0; j < 8; ++j)
            rowmin[(size_t)b * NPTS + n0 + 8 * hi + j] = rmin[j] + aa8[j];
    }
}

// Final reduction: out = (sum of all row-mins, both directions) / (B*N).
__global__ __launch_bounds__(256)
void chamfer_reduce(const float* __restrict__ mins, float* __restrict__ out,
                    int count, float scale)
{
    __shared__ float red[256];
    float s = 0.0f;
    for (int i = threadIdx.x; i < count; i += 256) s += mins[i];
    red[threadIdx.x] = s;
    __syncthreads();
    for (int off = 128; off > 0; off >>= 1) {
        if ((int)threadIdx.x < off) red[threadIdx.x] += red[threadIdx.x + off];
        __syncthreads();
    }
    if (threadIdx.x == 0) out[0] = red[0] * scale;
}

extern "C" void kernel_launch(void* const* d_in, const int* in_sizes, int n_in,
                              void* d_out, int out_size, void* d_ws, size_t ws_size,
                              hipStream_t stream) {
    (void)in_sizes; (void)n_in; (void)out_size; (void)ws_size;
    const float* tmpl = (const float*)d_in[0];   // [4, 4096, 3] fp32
    const float* src  = (const float*)d_in[1];   // [4, 4096, 3] fp32

    float* d01 = (float*)d_ws;                   // [B*N] template -> nearest source
    float* d10 = d01 + BATCH * NPTS;             // [B*M] source -> nearest template

    dim3 grid(NPTS / TILE / WAVES_PER_BLOCK, BATCH);

    chamfer_rowmin_wmma<<<grid, THREADS, 0, stream>>>(tmpl, src, d01);
    chamfer_rowmin_wmma<<<grid, THREADS, 0, stream>>>(src, tmpl, d10);

    chamfer_reduce<<<1, 256, 0, stream>>>(d01, (float*)d_out,
                                          2 * BATCH * NPTS,
                                          1.0f / (float)(BATCH * NPTS));
}